// GCNN_5978594476372
// MI455X (gfx1250) — compile-verified
//
#include <hip/hip_runtime.h>
#include <hip/hip_bf16.h>

typedef float v2f __attribute__((ext_vector_type(2)));
typedef float v8f __attribute__((ext_vector_type(8)));

#define IN_DIM 128
#define HID_DIM 128
#define OUT_DIM 40
#define BN_EPS 1e-5f

// ---------------------------------------------------------------------------
// Non-returning global float atomic add -> global_atomic_add_f32 (no RTN).
// Native f32 fadd is architecturally supported on CDNA5 (GLOBAL_ATOMIC_ADD_F32);
// avoid any CAS-loop fallback.
// ---------------------------------------------------------------------------
__device__ __forceinline__ void gatomic_fadd(float* p, float v) {
#if __has_builtin(__builtin_amdgcn_global_atomic_fadd_f32)
    (void)__builtin_amdgcn_global_atomic_fadd_f32(
        (__attribute__((address_space(1))) float*)p, v);
#else
    (void)__hip_atomic_fetch_add(p, v, __ATOMIC_RELAXED, __HIP_MEMORY_SCOPE_AGENT);
#endif
}

// ---------------------------------------------------------------------------
// Utility: fill buffer with a constant
// ---------------------------------------------------------------------------
__global__ void fill_f32(float* __restrict__ p, float v, long long n) {
    long long i = (long long)blockIdx.x * blockDim.x + threadIdx.x;
    if (i < n) p[i] = v;
}

// ---------------------------------------------------------------------------
// Degree accumulation: deg_out[src[e]] += 1 ; deg_in[dst[e]] += 1
// ---------------------------------------------------------------------------
__global__ void degree_kernel(const int* __restrict__ src, const int* __restrict__ dst,
                              float* __restrict__ dsrc, float* __restrict__ ddst,
                              long long E) {
    long long e = (long long)blockIdx.x * blockDim.x + threadIdx.x;
    if (e >= E) return;
    gatomic_fadd(&dsrc[src[e]], 1.0f);
    gatomic_fadd(&ddst[dst[e]], 1.0f);
}

// norm = rsqrt(max(deg, 1))
__global__ void finalize_norm(float* __restrict__ p, long long n) {
    long long i = (long long)blockIdx.x * blockDim.x + threadIdx.x;
    if (i < n) p[i] = rsqrtf(fmaxf(p[i], 1.0f));
}

// ---------------------------------------------------------------------------
// out[i,f] = x[i,f] * nsrc[i]   (row scaling, 128-wide rows)
// ---------------------------------------------------------------------------
__global__ void scale_rows(const float* __restrict__ x, const float* __restrict__ nsrc,
                           float* __restrict__ out, long long total) {
    long long i = (long long)blockIdx.x * blockDim.x + threadIdx.x;
    if (i >= total) return;
    out[i] = x[i] * nsrc[i >> 7];
}

// ---------------------------------------------------------------------------
// Edge scatter-add: one wave per edge, each lane moves 4 contiguous floats.
// agg[dst[e], :] += feat[src[e], :]
// Gathers are 512B-contiguous per edge; node features (51 MB) are L2-resident
// on MI455X (192 MB L2), so both gather and atomics run at L2 rate.
// ---------------------------------------------------------------------------
__global__ void edge_scatter(const float* __restrict__ feat,
                             const int* __restrict__ src, const int* __restrict__ dst,
                             float* __restrict__ agg, long long E) {
    long long tid = (long long)blockIdx.x * blockDim.x + threadIdx.x;
    long long e = tid >> 5;           // 32 lanes per edge
    if (e >= E) return;
    int c = (int)(tid & 31) * 4;      // feature chunk (4 floats)
    int s = src[e];
    int d = dst[e];
    const float4 v = *(const float4*)(feat + (size_t)s * 128 + c);
    float* p = agg + (size_t)d * 128 + c;
    gatomic_fadd(p + 0, v.x);
    gatomic_fadd(p + 1, v.y);
    gatomic_fadd(p + 2, v.z);
    gatomic_fadd(p + 3, v.w);
}

// ---------------------------------------------------------------------------
// WMMA f32 GEMM:  C[nrows x NCOLS] = (A * nd[row]) @ W[128 x NCOLS] + bias
// A is [nrows x 128] row-major. One wave per 16-row M tile; all N tiles of
// accumulators held in VGPRs. Uses V_WMMA_F32_16X16X4_F32 (exact f32 math).
//
// Fragment layouts (wave32, per CDNA5 ISA 7.12.2):
//  A 16x4:  lane&15 = M, lane>>4 selects K-pair; v0=K0/K2, v1=K1/K3
//  B 4x16:  lane&15 = N, lane>>4 selects K-pair; v0=K0/K2, v1=K1/K3
//  C 16x16: VGPR r -> row r (lanes 0-15) / row r+8 (lanes 16-31), N = lane&15
// ---------------------------------------------------------------------------
template <int NCOLS>
__global__ void gemm_wmma(const float* __restrict__ A, const float* __restrict__ nd,
                          const float* __restrict__ W, const float* __restrict__ bias,
                          float* __restrict__ C, int nrows) {
    constexpr int NT = (NCOLS + 15) / 16;
    const int lane = threadIdx.x & 31;
    const int wave = threadIdx.x >> 5;
    const int mtile = blockIdx.x * 4 + wave;      // blockDim.x == 128 -> 4 waves
    const int row0 = mtile * 16;
    if (row0 >= nrows) return;

    const int l15 = lane & 15;                    // M for A-frag / N for B-frag
    const int kh  = lane >> 4;                    // K-pair select

    int arow = row0 + l15;
    if (arow >= nrows) arow = nrows - 1;          // clamp (stores masked later)
    const float ndv = nd[arow];
    const float* __restrict__ Ap = A + (size_t)arow * 128;

    v8f acc[NT];
#pragma unroll
    for (int t = 0; t < NT; ++t) acc[t] = (v8f)(0.0f);

    for (int k0 = 0; k0 < 128; k0 += 4) {
        const int k = k0 + 2 * kh;
        v2f a;
        a.x = Ap[k + 0] * ndv;
        a.y = Ap[k + 1] * ndv;
#pragma unroll
        for (int t = 0; t < NT; ++t) {
            const int col = t * 16 + l15;
            v2f b;
            if ((NCOLS & 15) == 0 || col < NCOLS) {
                b.x = W[(size_t)(k + 0) * NCOLS + col];
                b.y = W[(size_t)(k + 1) * NCOLS + col];
            } else {
                b.x = 0.0f;
                b.y = 0.0f;
            }
            acc[t] = __builtin_amdgcn_wmma_f32_16x16x4_f32(
                /*neg_a=*/false, a, /*neg_b=*/false, b,
                /*c_mod=*/(short)0, acc[t], /*reuse_a=*/false, /*reuse_b=*/false);
        }
    }

    // Store D + bias
    const int rbase = row0 + 8 * kh;
#pragma unroll
    for (int t = 0; t < NT; ++t) {
        const int col = t * 16 + l15;
        if ((NCOLS & 15) == 0 || col < NCOLS) {
            const float bv = bias[col];
#pragma unroll
            for (int r = 0; r < 8; ++r) {
                const int row = rbase + r;
                if (row < nrows) C[(size_t)row * NCOLS + col] = acc[t][r] + bv;
            }
        }
    }
}

// ---------------------------------------------------------------------------
// BatchNorm statistics: per-feature sum and sum-of-squares.
// blockDim = 128 (thread == feature); 256 rows per block; one atomicAdd per
// thread per statistic.
// ---------------------------------------------------------------------------
__global__ void bn_stats(const float* __restrict__ h, float* __restrict__ sums,
                         float* __restrict__ sumsq, int nrows) {
    const int f = threadIdx.x;
    const int r0 = blockIdx.x * 256;
    int rend = r0 + 256;
    if (rend > nrows) rend = nrows;
    float s = 0.0f, ss = 0.0f;
    for (int r = r0; r < rend; ++r) {
        float v = h[(size_t)r * 128 + f];
        s += v;
        ss += v * v;
    }
    gatomic_fadd(&sums[f], s);
    gatomic_fadd(&sumsq[f], ss);
}

__global__ void bn_final(const float* __restrict__ sums, const float* __restrict__ sumsq,
                         const float* __restrict__ gamma, const float* __restrict__ beta,
                         float* __restrict__ scale, float* __restrict__ shift, int nrows) {
    const int f = threadIdx.x;   // 128 threads
    const float inv_n = 1.0f / (float)nrows;
    const float mean = sums[f] * inv_n;
    const float var = sumsq[f] * inv_n - mean * mean;
    const float istd = rsqrtf(var + BN_EPS);
    const float sc = gamma[f] * istd;
    scale[f] = sc;
    shift[f] = beta[f] - mean * sc;
}

// h[i,f] = max(0, h[i,f]*scale[f] + shift[f]) * nsrc[i]   (in place)
__global__ void bn_relu_scale(float* __restrict__ h, const float* __restrict__ scale,
                              const float* __restrict__ shift, const float* __restrict__ nsrc,
                              long long total) {
    long long i = (long long)blockIdx.x * blockDim.x + threadIdx.x;
    if (i >= total) return;
    const int f = (int)(i & 127);
    float v = h[i] * scale[f] + shift[f];
    v = fmaxf(v, 0.0f);
    h[i] = v * nsrc[i >> 7];
}

// ---------------------------------------------------------------------------
// Launch
// ---------------------------------------------------------------------------
extern "C" void kernel_launch(void* const* d_in, const int* in_sizes, int n_in,
                              void* d_out, int out_size, void* d_ws, size_t ws_size,
                              hipStream_t stream) {
    const float* x     = (const float*)d_in[0];
    const int*   src   = (const int*)d_in[1];
    const int*   dst   = (const int*)d_in[2];
    const float* W1    = (const float*)d_in[3];
    const float* b1    = (const float*)d_in[4];
    const float* gamma = (const float*)d_in[5];
    const float* beta  = (const float*)d_in[6];
    const float* W2    = (const float*)d_in[7];
    const float* b2    = (const float*)d_in[8];
    float* out = (float*)d_out;

    const long long N = in_sizes[0] / IN_DIM;   // 100000 nodes
    const long long E = in_sizes[1];            // 3200000 edges
    const long long NF = N * 128;

    // workspace carve-up (floats)
    float* ws    = (float*)d_ws;
    float* nsrc  = ws;                 // [N]  deg_out -> norm_src
    float* ndst  = ws + N;             // [N]  deg_in  -> norm_dst
    float* stats = ws + 2 * N;         // [512] sum|sumsq|scale|shift
    float* sums  = stats;
    float* sumsq = stats + 128;
    float* bscal = stats + 256;
    float* bshft = stats + 384;
    float* bufA  = stats + 512;        // [N*128] xs / h1 / h_scaled
    float* bufB  = bufA + NF;          // [N*128] agg

    const int TB = 256;
    auto blocks = [](long long n, int tb) { return (unsigned)((n + tb - 1) / tb); };

    // 1) degrees -> norms
    fill_f32<<<blocks(2 * N, TB), TB, 0, stream>>>(nsrc, 0.0f, 2 * N);
    degree_kernel<<<blocks(E, TB), TB, 0, stream>>>(src, dst, nsrc, ndst, E);
    finalize_norm<<<blocks(2 * N, TB), TB, 0, stream>>>(nsrc, 2 * N);

    // 2) layer 1: xs = x * nsrc ; agg = scatter_add(xs[src] -> dst)
    scale_rows<<<blocks(NF, TB), TB, 0, stream>>>(x, nsrc, bufA, NF);
    fill_f32<<<blocks(NF, TB), TB, 0, stream>>>(bufB, 0.0f, NF);
    edge_scatter<<<blocks(E * 32, TB), TB, 0, stream>>>(bufA, src, dst, bufB, E);

    // 3) h1 = (agg * ndst) @ W1 + b1   (WMMA f32)  -> bufA
    {
        const int mtiles = (int)((N + 15) / 16);
        gemm_wmma<HID_DIM><<<(mtiles + 3) / 4, 128, 0, stream>>>(bufB, ndst, W1, b1,
                                                                 bufA, (int)N);
    }

    // 4) batch norm (training-mode stats) + ReLU + norm_src scaling (in place)
    fill_f32<<<1, 256, 0, stream>>>(stats, 0.0f, 256);
    bn_stats<<<blocks(N, 256), 128, 0, stream>>>(bufA, sums, sumsq, (int)N);
    bn_final<<<1, 128, 0, stream>>>(sums, sumsq, gamma, beta, bscal, bshft, (int)N);
    bn_relu_scale<<<blocks(NF, TB), TB, 0, stream>>>(bufA, bscal, bshft, nsrc, NF);

    // 5) layer 2: agg2 = scatter_add(h_scaled[src] -> dst)
    fill_f32<<<blocks(NF, TB), TB, 0, stream>>>(bufB, 0.0f, NF);
    edge_scatter<<<blocks(E * 32, TB), TB, 0, stream>>>(bufA, src, dst, bufB, E);

    // 6) out = (agg2 * ndst) @ W2 + b2   (WMMA f32, 40 cols -> 3 N-tiles masked)
    {
        const int mtiles = (int)((N + 15) / 16);
        gemm_wmma<OUT_DIM><<<(mtiles + 3) / 4, 128, 0, stream>>>(bufB, ndst, W2, b2,
                                                                 out, (int)N);
    }
}